// SpikingSHDNet_RNN_50732153700703
// MI455X (gfx1250) — compile-verified
//
#include <hip/hip_runtime.h>
#include <math.h>

// Problem constants (from reference setup_inputs)
#define T_STEPS 16
#define NB      4096
#define DIN     700
#define CH      128
#define NCLS    20

#define NCq     (NB * CH)               // 524288  (one [N,C] plane)
#define TNC     (T_STEPS * NB * CH)     // 8388608 (one [T,N,C] tensor)

typedef float v2f __attribute__((ext_vector_type(2)));
typedef float v8f __attribute__((ext_vector_type(8)));

// Exact-fp32 tensor op: D = A(16x4) * B(4x16) + C(16x16)
__device__ __forceinline__ v8f wmma4(v2f a, v2f b, v8f c) {
    return __builtin_amdgcn_wmma_f32_16x16x4_f32(
        /*neg_a=*/false, a, /*neg_b=*/false, b,
        /*c_mod=*/(short)0, c, /*reuse_a=*/false, /*reuse_b=*/false);
}

// ---------------------------------------------------------------------------
// Kernel 1: pre[t*N+n][c] = x[n][t][:] @ W_in^T  + (b_in[c] + b_rec[c])
// One wave per 16x16 output tile; 8 waves/block share one LDS A-tile.
// K padded 700 -> 704, staged in 64-wide chunks.
// ---------------------------------------------------------------------------
#define LDSA_STR 68   // 16 rows * 68 floats; bank = (4m + k) mod 64 -> conflict-free
__global__ __launch_bounds__(256) void k_pre(const float* __restrict__ x,
                                             const float* __restrict__ Win,
                                             const float* __restrict__ bin,
                                             const float* __restrict__ brec,
                                             float* __restrict__ pre) {
    __shared__ float As[16 * LDSA_STR];
    const int tid  = threadIdx.x;
    const int lane = tid & 31;
    const int wave = tid >> 5;                 // 0..7 = column tile
    const int r0   = blockIdx.x * 16;          // global row base (r = t*N + n)
    const int t    = r0 / NB;
    const int n0   = r0 % NB;
    const int m    = lane & 15;
    const int g    = lane >> 4;
    const int ccol = wave * 16 + m;            // output/B column for this lane

    v8f acc = {0.f, 0.f, 0.f, 0.f, 0.f, 0.f, 0.f, 0.f};

    for (int k0 = 0; k0 < 704; k0 += 64) {
        __syncthreads();
        // stage 16 rows x 64 cols of A (zero-padded past K=700)
        #pragma unroll
        for (int j = 0; j < 4; ++j) {
            int e  = tid + 256 * j;
            int rr = e >> 6;
            int cc = e & 63;
            int k  = k0 + cc;
            float v = 0.f;
            if (k < DIN)
                v = x[((size_t)(n0 + rr) * T_STEPS + t) * DIN + k];
            As[rr * LDSA_STR + cc] = v;
        }
        __syncthreads();
        #pragma unroll
        for (int ks = 0; ks < 64; ks += 4) {
            const int kk = ks + 2 * g;         // K pair handled by this half-wave
            v2f a, b;
            a.x = As[m * LDSA_STR + kk];
            a.y = As[m * LDSA_STR + kk + 1];
            const int kg = k0 + kk;
            if (kg < DIN) {                    // A is zero there anyway; avoid OOB read
                b.x = Win[(size_t)ccol * DIN + kg];
                b.y = Win[(size_t)ccol * DIN + kg + 1];
            } else { b.x = 0.f; b.y = 0.f; }
            acc = wmma4(a, b, acc);
        }
    }
    const float bias = bin[ccol] + brec[ccol];
    #pragma unroll
    for (int j = 0; j < 8; ++j) {
        int mm = j + 8 * g;
        pre[(size_t)(r0 + mm) * CH + ccol] = acc[j] + bias;
    }
}

// ---------------------------------------------------------------------------
// Shared 128-K WMMA tile: A rows live in LDS (stride 132), W in global (L2).
// ---------------------------------------------------------------------------
#define LDSB_STR 132
__device__ __forceinline__ v8f gemm128_tile(const float* __restrict__ As,
                                            const float* __restrict__ W,
                                            int ccol, int m, int g) {
    v8f acc = {0.f, 0.f, 0.f, 0.f, 0.f, 0.f, 0.f, 0.f};
    #pragma unroll
    for (int ks = 0; ks < 128; ks += 4) {
        const int kk = ks + 2 * g;
        v2f a, b;
        a.x = As[m * LDSB_STR + kk];
        a.y = As[m * LDSB_STR + kk + 1];
        b.x = W[(size_t)ccol * CH + kk];
        b.y = W[(size_t)ccol * CH + kk + 1];
        acc = wmma4(a, b, acc);
    }
    return acc;
}

__device__ __forceinline__ void stage_rows(float* __restrict__ As,
                                           const float* __restrict__ src,
                                           int n0, int tid) {
    #pragma unroll
    for (int j = 0; j < 8; ++j) {
        int e  = tid + 256 * j;
        int rr = e >> 7;
        int cc = e & 127;
        As[rr * LDSB_STR + cc] = src[(size_t)(n0 + rr) * CH + cc];
    }
}

// ---------------------------------------------------------------------------
// Recurrent step, LIF #1:  cur = pre_t + s_prev @ W_rec^T ; v1,s1 update
// ---------------------------------------------------------------------------
__global__ __launch_bounds__(256) void k_step_rec(const float* __restrict__ pre_t,
                                                  const float* __restrict__ Wrec,
                                                  const float* __restrict__ sprev,
                                                  float* __restrict__ v1,
                                                  float* __restrict__ scur) {
    __shared__ float As[16 * LDSB_STR];
    const int tid = threadIdx.x, lane = tid & 31, wave = tid >> 5;
    const int n0 = blockIdx.x * 16;
    const int m = lane & 15, g = lane >> 4;
    const int ccol = wave * 16 + m;

    stage_rows(As, sprev, n0, tid);
    __syncthreads();
    v8f acc = gemm128_tile(As, Wrec, ccol, m, g);

    #pragma unroll
    for (int j = 0; j < 8; ++j) {
        int mm = j + 8 * g;
        size_t idx = (size_t)(n0 + mm) * CH + ccol;
        float cur = acc[j] + pre_t[idx];           // biases folded into pre
        float v   = 0.5f * (v1[idx] + cur);        // v += (x - v)/tau, tau=2
        float s   = (v >= 1.0f) ? 1.0f : 0.0f;     // spike, hard reset
        v1[idx]   = v * (1.0f - s);
        scur[idx] = s;
    }
}

// ---------------------------------------------------------------------------
// Feedforward, LIF #2:  x2 = s1 @ W2^T + b2 ; feat -> features[t]
// ---------------------------------------------------------------------------
__global__ __launch_bounds__(256) void k_step_ff(const float* __restrict__ s1,
                                                 const float* __restrict__ W2,
                                                 const float* __restrict__ b2,
                                                 float* __restrict__ v2,
                                                 float* __restrict__ feat_t) {
    __shared__ float As[16 * LDSB_STR];
    const int tid = threadIdx.x, lane = tid & 31, wave = tid >> 5;
    const int n0 = blockIdx.x * 16;
    const int m = lane & 15, g = lane >> 4;
    const int ccol = wave * 16 + m;

    stage_rows(As, s1, n0, tid);
    __syncthreads();
    v8f acc = gemm128_tile(As, W2, ccol, m, g);
    const float bias = b2[ccol];

    #pragma unroll
    for (int j = 0; j < 8; ++j) {
        int mm = j + 8 * g;
        size_t idx = (size_t)(n0 + mm) * CH + ccol;
        float v = 0.5f * (v2[idx] + acc[j] + bias);
        float s = (v >= 1.0f) ? 1.0f : 0.0f;
        v2[idx] = v * (1.0f - s);
        feat_t[idx] = s;
    }
}

// ---------------------------------------------------------------------------
// Scan 2, LIF #3:  h = attended_t @ W_p^T + b_p ; spike accumulated in ssum
// (outs.mean over T collapses to (mean_t s) @ W_out^T + b_out)
// ---------------------------------------------------------------------------
__global__ __launch_bounds__(256) void k_step2(const float* __restrict__ att_t,
                                               const float* __restrict__ Wp,
                                               const float* __restrict__ bp,
                                               float* __restrict__ v3,
                                               float* __restrict__ ssum) {
    __shared__ float As[16 * LDSB_STR];
    const int tid = threadIdx.x, lane = tid & 31, wave = tid >> 5;
    const int n0 = blockIdx.x * 16;
    const int m = lane & 15, g = lane >> 4;
    const int ccol = wave * 16 + m;

    stage_rows(As, att_t, n0, tid);
    __syncthreads();
    v8f acc = gemm128_tile(As, Wp, ccol, m, g);
    const float bias = bp[ccol];

    #pragma unroll
    for (int j = 0; j < 8; ++j) {
        int mm = j + 8 * g;
        size_t idx = (size_t)(n0 + mm) * CH + ccol;
        float v = 0.5f * (v3[idx] + acc[j] + bias);
        float s = (v >= 1.0f) ? 1.0f : 0.0f;
        v3[idx] = v * (1.0f - s);
        ssum[idx] += s;
    }
}

// --------------------------- elementwise helpers ---------------------------
__global__ void k_zero(float* p, int n) {
    int i = blockIdx.x * blockDim.x + threadIdx.x;
    if (i < n) p[i] = 0.f;
}

__global__ void k_meanC(const float* __restrict__ feats, float* __restrict__ mt) {
    int i = blockIdx.x * blockDim.x + threadIdx.x;   // i = t*N + n
    if (i >= T_STEPS * NB) return;
    const float* row = feats + (size_t)i * CH;
    float s = 0.f;
    #pragma unroll 8
    for (int c = 0; c < CH; ++c) s += row[c];
    mt[i] = s * (1.0f / CH);
}

__global__ void k_meanT(const float* __restrict__ feats, float* __restrict__ m0) {
    int i = blockIdx.x * blockDim.x + threadIdx.x;   // i = n*C + c
    if (i >= NCq) return;
    float s = 0.f;
    #pragma unroll
    for (int t = 0; t < T_STEPS; ++t) s += feats[(size_t)t * NCq + i];
    m0[i] = s * (1.0f / T_STEPS);
}

__device__ __forceinline__ float sigmoidf(float x) { return 1.0f / (1.0f + expf(-x)); }

__global__ void k_tatt(const float* __restrict__ mt, const float* __restrict__ kt,
                       float* __restrict__ tatt) {
    int i = blockIdx.x * blockDim.x + threadIdx.x;   // i = t*N + n
    if (i >= T_STEPS * NB) return;
    int t = i / NB, n = i % NB;
    float y = kt[1] * mt[i];
    if (t > 0)            y += kt[0] * mt[(t - 1) * NB + n];
    if (t < T_STEPS - 1)  y += kt[2] * mt[(t + 1) * NB + n];
    tatt[i] = sigmoidf(y);
}

__global__ void k_catt(const float* __restrict__ m0, const float* __restrict__ kc,
                       float* __restrict__ catt) {
    int i = blockIdx.x * blockDim.x + threadIdx.x;   // i = n*C + c
    if (i >= NCq) return;
    int c = i & (CH - 1);
    float y = kc[1] * m0[i];
    if (c > 0)      y += kc[0] * m0[i - 1];
    if (c < CH - 1) y += kc[2] * m0[i + 1];
    catt[i] = sigmoidf(y);
}

__global__ void k_attend(const float* __restrict__ feats, const float* __restrict__ tatt,
                         const float* __restrict__ catt, float* __restrict__ attn) {
    size_t i = (size_t)blockIdx.x * blockDim.x + threadIdx.x;  // over T*N*C
    if (i >= (size_t)TNC) return;
    size_t rem = i % (size_t)NCq;          // n*C + c
    int t = (int)(i / (size_t)NCq);
    int n = (int)(rem / CH);
    attn[i] = feats[i] * tatt[t * NB + n] * catt[rem];
}

__global__ void k_final(const float* __restrict__ ssum, const float* __restrict__ Wout,
                        const float* __restrict__ bout, float* __restrict__ out0) {
    int i = blockIdx.x * blockDim.x + threadIdx.x;   // i = n*NCLS + q
    if (i >= NB * NCLS) return;
    int n = i / NCLS, q = i % NCLS;
    const float* sr = ssum + (size_t)n * CH;
    const float* wr = Wout + (size_t)q * CH;
    float s = 0.f;
    #pragma unroll 8
    for (int k = 0; k < CH; ++k) s += sr[k] * wr[k];
    out0[i] = bout[q] + s * (1.0f / T_STEPS);
}

// ---------------------------------------------------------------------------
extern "C" void kernel_launch(void* const* d_in, const int* in_sizes, int n_in,
                              void* d_out, int out_size, void* d_ws, size_t ws_size,
                              hipStream_t stream) {
    const float* x    = (const float*)d_in[0];
    const float* Win  = (const float*)d_in[1];
    const float* bin  = (const float*)d_in[2];
    const float* Wrec = (const float*)d_in[3];
    const float* brec = (const float*)d_in[4];
    const float* W2   = (const float*)d_in[5];
    const float* b2   = (const float*)d_in[6];
    const float* kt   = (const float*)d_in[7];
    const float* kc   = (const float*)d_in[8];
    const float* Wp   = (const float*)d_in[9];
    const float* bp   = (const float*)d_in[10];
    const float* Wout = (const float*)d_in[11];
    const float* bout = (const float*)d_in[12];

    float* out0  = (float*)d_out;                    // [N, NCLS]
    float* feats = out0 + NB * NCLS;                 // [T, N, C]
    float* attn  = feats + (size_t)TNC;              // [T, N, C]

    float* ws    = (float*)d_ws;
    float* pre   = ws;                               // [T, N, C]
    float* v1    = pre + (size_t)TNC;
    float* v2    = v1 + NCq;
    float* v3    = v2 + NCq;
    float* sA    = v3 + NCq;
    float* sB    = sA + NCq;
    float* ssum  = sB + NCq;
    float* mt    = ssum + NCq;                       // [T, N]
    float* tatt  = mt + T_STEPS * NB;                // [T, N]
    float* m0    = tatt + T_STEPS * NB;              // [N, C]
    float* catt  = m0 + NCq;                         // [N, C]

    // zero v1,v2,v3,sA,sB,ssum (contiguous, 6 planes)
    {
        int n = 6 * NCq;
        k_zero<<<(n + 255) / 256, 256, 0, stream>>>(v1, n);
    }

    // hoisted input projection for all T (the dominant GEMM)
    k_pre<<<(T_STEPS * NB) / 16, 256, 0, stream>>>(x, Win, bin, brec, pre);

    // recurrent scan (stream-ordered dependency over t)
    for (int t = 0; t < T_STEPS; ++t) {
        float* sprev = (t & 1) ? sB : sA;
        float* scur  = (t & 1) ? sA : sB;
        k_step_rec<<<NB / 16, 256, 0, stream>>>(pre + (size_t)t * NCq, Wrec, sprev, v1, scur);
        k_step_ff <<<NB / 16, 256, 0, stream>>>(scur, W2, b2, v2, feats + (size_t)t * NCq);
    }

    // TATCA attention
    k_meanC <<<(T_STEPS * NB) / 256, 256, 0, stream>>>(feats, mt);
    k_meanT <<<NCq / 256, 256, 0, stream>>>(feats, m0);
    k_tatt  <<<(T_STEPS * NB) / 256, 256, 0, stream>>>(mt, kt, tatt);
    k_catt  <<<NCq / 256, 256, 0, stream>>>(m0, kc, catt);
    k_attend<<<TNC / 256, 256, 0, stream>>>(feats, tatt, catt, attn);

    // second scan (accumulate spikes; W_out applied once at the end)
    for (int t = 0; t < T_STEPS; ++t)
        k_step2<<<NB / 16, 256, 0, stream>>>(attn + (size_t)t * NCq, Wp, bp, v3, ssum);

    // classifier head: out = (ssum/T) @ W_out^T + b_out
    k_final<<<(NB * NCLS + 255) / 256, 256, 0, stream>>>(ssum, Wout, bout, out0);
}